// DGL_MPNNLayer_26465588478284
// MI455X (gfx1250) — compile-verified
//
#include <hip/hip_runtime.h>

// MPNN edge layer for MI455X (gfx1250, wave32).
//
// msg[e,o] = sum_{d,i} ef_ext[e,d] * x[e,i] * W'[d*32+i, o]
//   where ef_ext = [initial_ef[e,:], 1.0] (17 dims; the 1.0 row carries `be`),
//         W'[(d,i), o] = (d<16) ? We[d, i*32+o] : be[i*32+o]   (544 x 32, 68KB)
// => per 16-edge tile: msg(16x32) = Z(16x544) @ W'(544x32) via V_WMMA_F32_16X16X4_F32.
// Z is formed in-register (outer product ef x x); the lane-half K-split of the
// WMMA A operand is folded into the gather ADDRESS (lanes>=16 load x elements
// +2/+3 of each group of 4), so the hot loop has zero cndmask/select overhead.
// W' is staged once per block into LDS pre-swizzled into the WMMA B layout
// (one ds_load_b64 per WMMA). Scatter via global_atomic_add_f32.

typedef __attribute__((ext_vector_type(2))) float v2f;
typedef __attribute__((ext_vector_type(8))) float v8f;

#define N_NODES   50000
#define N_EDGES   200000
#define HID       32
#define EDGE_DIM  16
#define KTOT      ((EDGE_DIM + 1) * HID)   // 544
#define KSTEPS    (KTOT / 4)               // 136 WMMA K-steps
#define TILES     (N_EDGES / 16)           // 12500 (exact)
#define WPB       4                        // waves per block
#define BLOCKS    (TILES / WPB)            // 3125 (exact -> EXEC always all-1s)

__global__ void mpnn_init_out(const float* __restrict__ nf,
                              const float* __restrict__ bias,
                              float* __restrict__ out) {
    int idx = blockIdx.x * blockDim.x + threadIdx.x;
    if (idx < N_NODES * HID) {
        out[idx] = nf[idx] + bias[idx & (HID - 1)];
    }
}

__launch_bounds__(32 * WPB, 1)
__global__ void mpnn_edge_wmma(const float* __restrict__ nf,
                               const float* __restrict__ ef,
                               const int*   __restrict__ src,
                               const int*   __restrict__ dst,
                               const float* __restrict__ We,
                               const float* __restrict__ be,
                               float* __restrict__ out) {
    // B operand for the whole K range, pre-swizzled into WMMA B layout:
    // entry (kk, ntile, lane): {W'[kk*4+koff, n], W'[kk*4+koff+1, n]}
    //   koff = (lane>=16) ? 2 : 0,  n = ntile*16 + (lane&15)
    __shared__ v2f lds_b[KSTEPS * 2 * 32];   // 69632 bytes

    for (int idx = threadIdx.x; idx < KSTEPS * 2 * 32; idx += blockDim.x) {
        const int lane = idx & 31;
        const int nt   = (idx >> 5) & 1;
        const int kk   = idx >> 6;
        const int koff = (lane & 16) ? 2 : 0;
        const int n    = nt * 16 + (lane & 15);
        const int k0   = kk * 4 + koff;
        v2f v;
        {
            const int k = k0, d = k >> 5, i = k & 31;
            v.x = (d < EDGE_DIM) ? We[d * (HID * HID) + i * HID + n]
                                 : be[i * HID + n];
        }
        {
            const int k = k0 + 1, d = k >> 5, i = k & 31;
            v.y = (d < EDGE_DIM) ? We[d * (HID * HID) + i * HID + n]
                                 : be[i * HID + n];
        }
        lds_b[idx] = v;
    }
    __syncthreads();

    const int  lane = threadIdx.x & 31;
    const bool hi   = (lane & 16) != 0;       // high half-wave -> K offsets +2/+3
    const int  wave = threadIdx.x >> 5;
    const int  tile = blockIdx.x * WPB + wave; // always < TILES (grid is exact)
    const int  e    = tile * 16 + (lane & 15); // edge handled by this lane (row M)
    const int  s    = src[e];
    const int  dn   = dst[e];

    // Gather ONLY the x elements this lane feeds into WMMA A: for group g of 4
    // hidden dims, lanes<16 need {4g,4g+1}, lanes>=16 need {4g+2,4g+3}. The
    // half-wave select is folded into the base address (no per-lane register
    // indexing -> no cndmask trees).
    const float* xbase = nf + (size_t)s * HID + (hi ? 2 : 0);
    float2 x2[HID / 4];
    #pragma unroll
    for (int g = 0; g < HID / 4; ++g) {
        x2[g] = *reinterpret_cast<const float2*>(xbase + 4 * g);  // global_load_b64
    }

    float efv[EDGE_DIM + 1];
    {
        const float4* ep = reinterpret_cast<const float4*>(ef + (size_t)e * EDGE_DIM);
        #pragma unroll
        for (int q = 0; q < EDGE_DIM / 4; ++q) {
            float4 t = ep[q];
            efv[4 * q + 0] = t.x; efv[4 * q + 1] = t.y;
            efv[4 * q + 2] = t.z; efv[4 * q + 3] = t.w;
        }
    }
    efv[EDGE_DIM] = 1.0f;  // virtual edge-dim: selects the bias rows of W'

    v8f acc0 = {};  // output columns  0..15
    v8f acc1 = {};  // output columns 16..31

    #pragma unroll
    for (int kk = 0; kk < KSTEPS; ++kk) {
        const float  ed = efv[kk >> 3];     // constant index after unroll
        const float2 xs = x2[kk & 7];       // constant index after unroll
        v2f a; a.x = ed * xs.x; a.y = ed * xs.y;   // one v_dual_mul
        v2f b0 = lds_b[(kk * 2 + 0) * 32 + lane];
        v2f b1 = lds_b[(kk * 2 + 1) * 32 + lane];
        acc0 = __builtin_amdgcn_wmma_f32_16x16x4_f32(
            false, a, false, b0, (short)0, acc0, false, false);
        acc1 = __builtin_amdgcn_wmma_f32_16x16x4_f32(
            false, a, false, b1, (short)0, acc1, false, false);
    }

    // C/D layout: VGPR j holds (M = j + (hi?8:0), N = lane&15). Scatter-add.
    const int o0 = lane & 15;
    #pragma unroll
    for (int j = 0; j < 8; ++j) {
        const int m    = j + (hi ? 8 : 0);       // edge-in-tile owning this row
        const int node = __shfl(dn, m, 16);      // dst node of that edge (ds_bpermute)
        float* row = out + (size_t)node * HID;
        atomicAdd(row + o0,      acc0[j]);
        atomicAdd(row + 16 + o0, acc1[j]);
    }
}

extern "C" void kernel_launch(void* const* d_in, const int* in_sizes, int n_in,
                              void* d_out, int out_size, void* d_ws, size_t ws_size,
                              hipStream_t stream) {
    const float* nf   = (const float*)d_in[0];
    const float* ef   = (const float*)d_in[1];
    const int*   src  = (const int*)  d_in[2];
    const int*   dst  = (const int*)  d_in[3];
    const float* We   = (const float*)d_in[4];
    const float* be   = (const float*)d_in[5];
    const float* bias = (const float*)d_in[6];
    float* out = (float*)d_out;

    // out = nf + bias (also provides the deterministic zero-base for atomics)
    {
        const int total = N_NODES * HID;
        mpnn_init_out<<<(total + 255) / 256, 256, 0, stream>>>(nf, bias, out);
    }
    // fused edge GEMM + scatter
    mpnn_edge_wmma<<<BLOCKS, 32 * WPB, 0, stream>>>(nf, ef, src, dst, We, be, out);
    (void)d_ws; (void)ws_size; (void)in_sizes; (void)n_in; (void)out_size;
}